// YOLOLoss_17901423689891
// MI455X (gfx1250) — compile-verified
//
#include <hip/hip_runtime.h>
#include <math.h>

#define NUM_CLASSES 80
#define IOU_THRESH   0.213f
#define IGNORE_THRESH 0.7f
#define COORD_SCALE  0.07f

typedef float v2f __attribute__((ext_vector_type(2)));
typedef float v8f __attribute__((ext_vector_type(8)));

// anchors / reduction per level (BASE_ANCHORS[mask]/reduction)
__device__ const float g_anch[3][3][2] = {
  {{12.f/ 8.f, 16.f/ 8.f}, {19.f/ 8.f, 36.f/ 8.f}, {40.f/ 8.f,  28.f/ 8.f}},
  {{36.f/16.f, 75.f/16.f}, {76.f/16.f, 55.f/16.f}, {72.f/16.f, 146.f/16.f}},
  {{142.f/32.f,110.f/32.f},{192.f/32.f,243.f/32.f},{459.f/32.f, 401.f/32.f}},
};
__device__ const float g_red[3] = {8.f, 16.f, 32.f};

__device__ __forceinline__ float sigmoidf_(float x) { return 1.f / (1.f + expf(-x)); }
// logaddexp(x,0) computed stably
__device__ __forceinline__ float softplusf_(float x) {
  return fmaxf(x, 0.f) + log1pf(expf(-fabsf(x)));
}

__device__ __forceinline__ float ciou_f(float gx, float gy, float gw, float gh,
                                        float tx, float ty, float tw, float th) {
  float gx1 = gx - gw * 0.5f, gx2 = gx + gw * 0.5f;
  float gy1 = gy - gh * 0.5f, gy2 = gy + gh * 0.5f;
  float tx1 = tx - tw * 0.5f, tx2 = tx + tw * 0.5f;
  float ty1 = ty - th * 0.5f, ty2 = ty + th * 0.5f;
  float iw = fmaxf(fminf(gx2, tx2) - fmaxf(gx1, tx1), 0.f);
  float ih = fmaxf(fminf(gy2, ty2) - fmaxf(gy1, ty1), 0.f);
  float inter = iw * ih;
  float uni = gw * gh + tw * th + 1e-16f - inter;
  float iou = inter / uni;
  float cw = fmaxf(gx2, tx2) - fminf(gx1, tx1);
  float ch = fmaxf(gy2, ty2) - fminf(gy1, ty1);
  float c2 = cw * cw + ch * ch + 1e-16f;
  float rho2 = (gx - tx) * (gx - tx) + (gy - ty) * (gy - ty);
  const float K = 0.405284734569351f;  // 4/pi^2
  float dv = atanf(tw / th) - atanf(gw / gh);
  float v = K * dv * dv;
  float alpha = v / (1.f - iou + v + 1e-16f);
  return iou - (rho2 / c2 + v * alpha);
}

// ---- kernel 0: per-level target metadata ----------------------------------
// ws float layout per level L (base L*512):
//   [0..63] cx  [64..127] cy  [128..191] w  [192..255] h
//   ints: [256..319] ti  [320..383] tj  [384..447] bid  [448..511] match-mask
__global__ void setup_meta(const float* __restrict__ targets, float* __restrict__ wsF) {
  int t = threadIdx.x;
  if (t >= 192) return;
  int L = t / 64, m = t % 64;
  float r  = g_red[L];
  float x1 = targets[m * 5 + 0] / r, y1 = targets[m * 5 + 1] / r;
  float x2 = targets[m * 5 + 2] / r, y2 = targets[m * 5 + 3] / r;
  int   bid = (int)targets[m * 5 + 4];
  float w = x2 - x1, h = y2 - y1;
  float cx = (x1 + x2) * 0.5f, cy = (y1 + y2) * 0.5f;
  int ti = (int)cx, tj = (int)cy;   // positive -> trunc == astype(int32)
  float tarea = w * h;
  float io[3]; float best = -1.f; int bi = 0;
  for (int a = 0; a < 3; ++a) {
    float aw = g_anch[L][a][0], ah = g_anch[L][a][1];
    float inter = fminf(aw, w) * fminf(ah, h);
    io[a] = inter / (aw * ah + tarea - inter);
    if (io[a] > best) { best = io[a]; bi = a; }   // argmax picks first on ties
  }
  int mask = 0;
  for (int a = 0; a < 3; ++a)
    if (io[a] > IOU_THRESH || a == bi) mask |= (1 << a);
  float* fb = wsF + L * 512;
  int*   ib = (int*)fb;
  fb[m] = cx; fb[64 + m] = cy; fb[128 + m] = w; fb[192 + m] = h;
  ib[256 + m] = ti; ib[320 + m] = tj; ib[384 + m] = bid; ib[448 + m] = mask;
}

// ---- kernel 1: matched-target losses (ciou + obj_pos + cls) ---------------
__global__ void target_loss(const float* __restrict__ pred, const int* __restrict__ cats,
                            const float* __restrict__ wsF, int level, int G, float xys,
                            float* __restrict__ partials, int nslot, int slot) {
  int t = threadIdx.x;
  float iou_c = 0.f, obj_c = 0.f, cls_c = 0.f;
  if (t < 192) {
    int a = t / 64, m = t % 64;
    const float* fb = wsF + level * 512;
    const int*   ib = (const int*)fb;
    int mask = ib[448 + m];
    if (mask & (1 << a)) {
      float cx = fb[m], cy = fb[64 + m], w = fb[128 + m], h = fb[192 + m];
      int ti = ib[256 + m], tj = ib[320 + m], b = ib[384 + m];
      long HW = (long)G * G;
      const float* base = pred + (long)(b * 255 + a * 85) * HW + (long)tj * G + ti;
      float p0 = base[0], p1 = base[HW], p2 = base[2 * HW], p3 = base[3 * HW], p4 = base[4 * HW];
      float half = 0.5f * (xys - 1.f);
      float gx = sigmoidf_(p0) * xys - half;
      float gy = sigmoidf_(p1) * xys - half;
      float gw = expf(p2) * g_anch[level][a][0];
      float gh = expf(p3) * g_anch[level][a][1];
      float tx = cx - (float)ti, ty = cy - (float)tj;
      iou_c = 1.f - ciou_f(gx, gy, gw, gh, tx, ty, w, h);
      obj_c = softplusf_(p4) - p4;                 // bce(logit, 1)
      __builtin_prefetch(base + 5 * HW, 0, 1);     // global_prefetch_b8
      float sp = 0.f;
      for (int c = 0; c < NUM_CLASSES; ++c) sp += softplusf_(base[(long)(5 + c) * HW]);
      cls_c = sp - base[(long)(5 + cats[m] - 1) * HW];  // sum bce vs one-hot
    }
  }
  __shared__ float r0[256], r1[256], r2[256];
  r0[t] = iou_c; r1[t] = obj_c; r2[t] = cls_c;
  __syncthreads();
  for (int s = 128; s > 0; s >>= 1) {
    if (t < s) { r0[t] += r0[t + s]; r1[t] += r1[t + s]; r2[t] += r2[t + s]; }
    __syncthreads();
  }
  if (t == 0) {
    partials[slot]             = r0[0];
    partials[nslot + slot]     = r1[0];
    partials[2 * nslot + slot] = r2[0];
  }
}

// ---- kernel 2: grid obj_neg (ignore mask + matched-cell exclusion) --------
__global__ void grid_obj(const float* __restrict__ pred, const float* __restrict__ wsF,
                         int level, int G, float xys,
                         float* __restrict__ partials, int nslot, int slotBase) {
  __shared__ float s_cx[64], s_cy[64], s_w[64], s_h[64];
  __shared__ int   s_ti[64], s_tj[64], s_bid[64], s_mask[64];
  __shared__ float red[256];
  int tid = threadIdx.x;
  if (tid < 64) {
    const float* fb = wsF + level * 512;
    const int*   ib = (const int*)fb;
    s_cx[tid] = fb[tid];       s_cy[tid]  = fb[64 + tid];
    s_w[tid]  = fb[128 + tid]; s_h[tid]   = fb[192 + tid];
    s_ti[tid] = ib[256 + tid]; s_tj[tid]  = ib[320 + tid];
    s_bid[tid]= ib[384 + tid]; s_mask[tid]= ib[448 + tid];
  }
  __syncthreads();
  int HW = G * G;
  int Npos = 8 * 3 * HW;
  int t = blockIdx.x * blockDim.x + tid;
  float contrib = 0.f;
  if (t < Npos) {
    int i = t % G;
    int j = (t / G) % G;
    int a = (t / HW) % 3;
    int b =  t / (HW * 3);
    const float* base = pred + (long)(b * 255 + a * 85) * HW + (long)j * G + i;
    float p0 = base[0], p1 = base[HW], p2 = base[2 * HW], p3 = base[3 * HW], p4 = base[4 * HW];
    float half = 0.5f * (xys - 1.f);
    float fx = sigmoidf_(p0) * xys - half + (float)i;
    float fy = sigmoidf_(p1) * xys - half + (float)j;
    float fw = expf(p2) * g_anch[level][a][0];
    float fh = expf(p3) * g_anch[level][a][1];
    float fx1 = fx - fw * 0.5f, fx2 = fx + fw * 0.5f;
    float fy1 = fy - fh * 0.5f, fy2 = fy + fh * 0.5f;
    float farea = fw * fh;
    bool zero = false;
    for (int m = 0; m < 64; ++m) {
      if (s_bid[m] != b) continue;
      if (s_tj[m] == j && s_ti[m] == i && (s_mask[m] & (1 << a))) zero = true;
      float tx1 = s_cx[m] - s_w[m] * 0.5f, tx2 = s_cx[m] + s_w[m] * 0.5f;
      float ty1 = s_cy[m] - s_h[m] * 0.5f, ty2 = s_cy[m] + s_h[m] * 0.5f;
      float iw = fmaxf(fminf(fx2, tx2) - fmaxf(fx1, tx1), 0.f);
      float ih = fmaxf(fminf(fy2, ty2) - fmaxf(fy1, ty1), 0.f);
      float inter = iw * ih;
      float uni = farea + s_w[m] * s_h[m] + 1e-16f - inter;
      if (inter / uni > IGNORE_THRESH) zero = true;
    }
    if (!zero) contrib = softplusf_(p4);
  }
  red[tid] = contrib;
  __syncthreads();
  for (int s = 128; s > 0; s >>= 1) {
    if (tid < s) red[tid] += red[tid + s];
    __syncthreads();
  }
  if (tid == 0) {
    int slot = slotBase + blockIdx.x;
    partials[slot]             = 0.f;
    partials[nslot + slot]     = red[0];
    partials[2 * nslot + slot] = 0.f;
  }
}

// ---- kernel 3: single-wave f32 WMMA reduction -----------------------------
// A (16x4 f32): lane L puts its accumulator in A-vgpr0, 0 in A-vgpr1
//   -> A[m,0]=acc_m, A[m,2]=acc_{m+16}, A[m,1]=A[m,3]=0
// B = all ones -> D[m,n] = acc_m + acc_{m+16} for every n.
// lane n<16 holds D[0..7,n] in its 8 C vgprs; lane n>=16 holds D[8..15,n-16].
// total = (sum of own 8) + shfl_xor(.,16).
__global__ void reduce_final(const float* __restrict__ partials, int nslot,
                             float* __restrict__ out) {
  int lane = threadIdx.x;   // 32 lanes, full wave, EXEC all ones
  float acc[3] = {0.f, 0.f, 0.f};
  for (int s = lane; s < nslot; s += 32) {
    acc[0] += partials[s];
    acc[1] += partials[nslot + s];
    acc[2] += partials[2 * nslot + s];
  }
  v2f onesB; onesB[0] = 1.f; onesB[1] = 1.f;
  float tot[3];
  for (int c = 0; c < 3; ++c) {
    v2f a; a[0] = acc[c]; a[1] = 0.f;
    v8f cz = {};
    v8f d = __builtin_amdgcn_wmma_f32_16x16x4_f32(
        /*neg_a=*/false, a, /*neg_b=*/false, onesB,
        /*c_mod=*/(short)0, cz, /*reuse_a=*/false, /*reuse_b=*/false);
    float s = 0.f;
    for (int r = 0; r < 8; ++r) s += d[r];
    tot[c] = s + __shfl_xor(s, 16, 32);
  }
  if (lane == 0) {
    out[0] = tot[0] * COORD_SCALE;  // iou loss
    out[1] = tot[1];                // obj loss (OBJ_SCALE = NOOBJ_SCALE = 1)
    out[2] = tot[2];                // cls loss (CLASS_SCALE = 1)
  }
}

extern "C" void kernel_launch(void* const* d_in, const int* in_sizes, int n_in,
                              void* d_out, int out_size, void* d_ws, size_t ws_size,
                              hipStream_t stream) {
  const float* preds[3] = {(const float*)d_in[0], (const float*)d_in[1], (const float*)d_in[2]};
  const float* targets  = (const float*)d_in[3];
  const int*   cats     = (const int*)d_in[4];
  float* out = (float*)d_out;
  float* wsF = (float*)d_ws;
  float* partials = wsF + 3 * 512;  // after per-level metadata

  const int   Gs[3]  = {76, 38, 19};
  const float xys[3] = {1.2f, 1.1f, 1.05f};

  int nb[3];
  for (int L = 0; L < 3; ++L) {
    int npos = 8 * 3 * Gs[L] * Gs[L];
    nb[L] = (npos + 255) / 256;
  }
  int nslot = 3 + nb[0] + nb[1] + nb[2];

  setup_meta<<<1, 192, 0, stream>>>(targets, wsF);
  for (int L = 0; L < 3; ++L)
    target_loss<<<1, 256, 0, stream>>>(preds[L], cats, wsF, L, Gs[L], xys[L],
                                       partials, nslot, L);
  int slotBase = 3;
  for (int L = 0; L < 3; ++L) {
    grid_obj<<<nb[L], 256, 0, stream>>>(preds[L], wsF, L, Gs[L], xys[L],
                                        partials, nslot, slotBase);
    slotBase += nb[L];
  }
  reduce_final<<<1, 32, 0, stream>>>(partials, nslot, out);
}